// ODEfunc_34943853920779
// MI455X (gfx1250) — compile-verified
//
#include <hip/hip_runtime.h>
#include <stdint.h>

// Lorenz tangent-map RHS:  out[:,0:3] = f(x),  out[:,3:12] = (A(x) @ Phi) row-major
//   A = [[-s,  s,  0],
//        [r-x3,-1,-x1],
//        [x2, x1, -b]]
// Memory-bound streaming kernel: 1.536 GB total @ 23.3 TB/s -> ~66 us floor.
// CDNA5 path: async global<->LDS staging (512B coalesced per instruction,
// ASYNCcnt-tracked), wave-private tiles => no barriers. Stream is touch-once,
// so both directions use non-temporal hints (192MB L2 << 1.5GB working set).

#define L_SIGMA 10.0f
#define L_RHO   28.0f
#define L_BETA  (8.0f / 3.0f)

constexpr int TPB = 256;                 // 8 waves of 32 on wave32 CDNA5
constexpr int PARTICLES_PER_BLOCK = TPB; // 1 particle / thread
constexpr int BYTES_PER_WAVE = 32 * 48;  // 1536 B tile per wave
constexpr int FLOATS_PER_BLOCK = PARTICLES_PER_BLOCK * 12; // 3072 floats = 12 KB LDS

__device__ __forceinline__ void lorenz_rhs(const float4 a, const float4 b, const float4 c,
                                           float4& o0, float4& o1, float4& o2) {
  const float x1 = a.x, x2 = a.y, x3 = a.z;
  const float P00 = a.w, P01 = b.x, P02 = b.y;
  const float P10 = b.z, P11 = b.w, P12 = c.x;
  const float P20 = c.y, P21 = c.z, P22 = c.w;

  const float d0 = L_SIGMA * (x2 - x1);
  const float d1 = fmaf(L_RHO, x1, fmaf(-x1, x3, -x2));
  const float d2 = fmaf(x1, x2, -L_BETA * x3);

  const float a10 = L_RHO - x3;
  // row 0 of Phi_dot:  s*(P1j - P0j)
  const float r00 = L_SIGMA * (P10 - P00);
  const float r01 = L_SIGMA * (P11 - P01);
  const float r02 = L_SIGMA * (P12 - P02);
  // row 1: (r-x3)*P0j - P1j - x1*P2j
  const float r10 = fmaf(a10, P00, fmaf(-x1, P20, -P10));
  const float r11 = fmaf(a10, P01, fmaf(-x1, P21, -P11));
  const float r12 = fmaf(a10, P02, fmaf(-x1, P22, -P12));
  // row 2: x2*P0j + x1*P1j - b*P2j
  const float r20 = fmaf(x2, P00, fmaf(x1, P10, -L_BETA * P20));
  const float r21 = fmaf(x2, P01, fmaf(x1, P11, -L_BETA * P21));
  const float r22 = fmaf(x2, P02, fmaf(x1, P12, -L_BETA * P22));

  o0 = make_float4(d0, d1, d2, r00);
  o1 = make_float4(r01, r02, r10, r11);
  o2 = make_float4(r12, r20, r21, r22);
}

__global__ void __launch_bounds__(TPB)
lorenz_tangent_kernel(const float* __restrict__ y, float* __restrict__ out, int n) {
  __shared__ __align__(16) float sh[FLOATS_PER_BLOCK];

  const int lane = threadIdx.x & 31;
  const int wave = threadIdx.x >> 5;
  const long long blockBase = (long long)blockIdx.x * PARTICLES_PER_BLOCK;

  if (blockBase + PARTICLES_PER_BLOCK <= (long long)n) {
    // ---- fast path: async staged, wave-private 1536B tiles, no barriers ----
    // Low 32 bits of the flat shared-aperture address == LDS byte offset.
    const unsigned shBase  = (unsigned)(uintptr_t)(void*)sh;
    // One LDS-addr VGPR + one 64-bit global addr pair per direction; the
    // three 512B chunks differ only by the IOFFSET immediate (ISA: offset is
    // added to BOTH the LDS and the global address for async LDS ops).
    const unsigned l = shBase + (unsigned)(wave * BYTES_PER_WAVE + lane * 16);
    const uint64_t gIn = (uint64_t)(uintptr_t)y + (uint64_t)blockBase * 48ull
                         + (uint64_t)(wave * BYTES_PER_WAVE + lane * 16);

    asm volatile("global_load_async_to_lds_b128 %0, %1, off th:TH_LOAD_NT"
                 :: "v"(l), "v"(gIn) : "memory");
    asm volatile("global_load_async_to_lds_b128 %0, %1, off offset:512 th:TH_LOAD_NT"
                 :: "v"(l), "v"(gIn) : "memory");
    asm volatile("global_load_async_to_lds_b128 %0, %1, off offset:1024 th:TH_LOAD_NT"
                 :: "v"(l), "v"(gIn) : "memory");
    asm volatile("s_wait_asynccnt 0" ::: "memory");

    // Each lane reads its own particle's 48B (16B-aligned) from the wave tile.
    const float4* sp = (const float4*)(sh + wave * 384 + lane * 12);
    const float4 a = sp[0], b = sp[1], c = sp[2];

    float4 o0, o1, o2;
    lorenz_rhs(a, b, c, o0, o1, o2);

    // Write results back into the same tile (per-wave in-order DS makes the
    // read-before-overwrite safe; same addresses alias anyway).
    float4* spw = (float4*)(sh + wave * 384 + lane * 12);
    spw[0] = o0; spw[1] = o1; spw[2] = o2;
    asm volatile("s_wait_dscnt 0" ::: "memory");

    const uint64_t gOut = (uint64_t)(uintptr_t)out + (uint64_t)blockBase * 48ull
                          + (uint64_t)(wave * BYTES_PER_WAVE + lane * 16);
    asm volatile("global_store_async_from_lds_b128 %0, %1, off th:TH_STORE_NT"
                 :: "v"(gOut), "v"(l) : "memory");
    asm volatile("global_store_async_from_lds_b128 %0, %1, off offset:512 th:TH_STORE_NT"
                 :: "v"(gOut), "v"(l) : "memory");
    asm volatile("global_store_async_from_lds_b128 %0, %1, off offset:1024 th:TH_STORE_NT"
                 :: "v"(gOut), "v"(l) : "memory");
    asm volatile("s_wait_asynccnt 0" ::: "memory");
  } else {
    // ---- tail path: direct vectorized loads/stores with bounds check ----
    const int t = (int)blockBase + (int)threadIdx.x;
    if (t < n) {
      const float4* yp = (const float4*)(y + (size_t)t * 12);
      const float4 a = yp[0], b = yp[1], c = yp[2];
      float4 o0, o1, o2;
      lorenz_rhs(a, b, c, o0, o1, o2);
      float4* op = (float4*)(out + (size_t)t * 12);
      op[0] = o0; op[1] = o1; op[2] = o2;
    }
  }
}

extern "C" void kernel_launch(void* const* d_in, const int* in_sizes, int n_in,
                              void* d_out, int out_size, void* d_ws, size_t ws_size,
                              hipStream_t stream) {
  // inputs: d_in[0] = t (1 float, unused by the RHS), d_in[1] = y (N*12 f32)
  const float* y = (const float*)d_in[1];
  float* out = (float*)d_out;
  const int n = in_sizes[1] / 12;
  if (n <= 0) return;
  const int blocks = (n + TPB - 1) / TPB;
  lorenz_tangent_kernel<<<blocks, TPB, 0, stream>>>(y, out, n);
}